// SimpleMoEClassifier_86681029968546
// MI455X (gfx1250) — compile-verified
//
#include <hip/hip_runtime.h>
#include <cstdint>

#define N_EXPERTS 23
#define IN_DIM    267
#define HID       1024
#define N_LAYERS  4
#define N_CLASSES 5
#define TOP_K     2
#define BATCH     1024

#define CAP        4096          // padded permuted-row capacity (>= 2048 + 23*63)
#define MTILE      64            // rows per M tile (segments padded to this)
#define MAX_MTILES (CAP / MTILE)
#define KCHUNK     32            // K staged through LDS per iteration
#define XG_STRIDE  288           // IN_DIM rounded up to multiple of KCHUNK
#define LDS_STRIDE 36            // padded K-stride in LDS (bank-conflict free, 16B-aligned rows)

static_assert(CAP % MTILE == 0, "");
static_assert(XG_STRIDE % KCHUNK == 0 && XG_STRIDE >= IN_DIM, "");
static_assert((LDS_STRIDE * 4) % 16 == 0, "LDS rows must stay 16B aligned for B128 copies");

typedef __attribute__((ext_vector_type(2))) float v2f;
typedef __attribute__((ext_vector_type(8))) float v8f;
typedef int v4i_ __attribute__((vector_size(16)));   // matches builtin's expected element type

// ---- CDNA5 async global->LDS path (ASYNCcnt), guarded so either toolchain compiles ----
#if defined(__has_builtin)
#if __has_builtin(__builtin_amdgcn_global_load_async_to_lds_b128)
#define HAS_ASYNC_LDS 1
#else
#define HAS_ASYNC_LDS 0
#endif
#else
#define HAS_ASYNC_LDS 0
#endif

#define AS1 __attribute__((address_space(1)))
#define AS3 __attribute__((address_space(3)))

__device__ __forceinline__ void async_copy16(const float* g, float* l) {
#if HAS_ASYNC_LDS
    __builtin_amdgcn_global_load_async_to_lds_b128((AS1 v4i_*)g, (AS3 v4i_*)l, 0, 0);
#else
    (void)g; (void)l;
#endif
}

#if defined(__has_builtin) && __has_builtin(__builtin_amdgcn_s_wait_asynccnt)
#define WAIT_ASYNC(n) __builtin_amdgcn_s_wait_asynccnt(n)
#else
#define WAIT_ASYNC(n) asm volatile("s_wait_asynccnt %0" ::"i"(n) : "memory")
#endif

__device__ __forceinline__ float gelu_erf(float v) {
    return 0.5f * v * (1.0f + erff(v * 0.7071067811865475f));
}
__device__ __forceinline__ float wave_max32(float v) {
    for (int o = 16; o > 0; o >>= 1) v = fmaxf(v, __shfl_xor(v, o, 32));
    return v;
}
__device__ __forceinline__ float wave_sum32(float v) {
    for (int o = 16; o > 0; o >>= 1) v += __shfl_xor(v, o, 32);
    return v;
}

// ---------------- setup kernels ----------------

__global__ void init_kernel(int* counts, int* cursor) {
    int i = threadIdx.x;
    if (i < N_EXPERTS) { counts[i] = 0; cursor[i] = 0; }
}

// one wave per token: logits over 23 experts, softmax, top-2, renormalize
__global__ void router_kernel(const float* __restrict__ x,
                              const float* __restrict__ rw,
                              const float* __restrict__ rb,
                              int* __restrict__ topk_idx,
                              float* __restrict__ topk_w,
                              int* __restrict__ counts) {
    int b = blockIdx.x;
    int lane = threadIdx.x;
    float logit = -3.4e38f;
    if (lane < N_EXPERTS) {
        float s = rb[lane];
        const float* xr = x + (size_t)b * IN_DIM;
        const float* wr = rw + (size_t)lane * IN_DIM;
        for (int k = 0; k < IN_DIM; ++k) s += xr[k] * wr[k];
        logit = s;
    }
    float mx = wave_max32(logit);
    float p = (lane < N_EXPERTS) ? expf(logit - mx) : 0.0f;
    float den = wave_sum32(p);
    p /= den;
    __shared__ float sp[32];
    sp[lane] = p;
    __syncthreads();
    if (lane == 0) {
        int i0 = -1, i1 = -1; float p0 = -1.0f, p1 = -1.0f;
        for (int e = 0; e < N_EXPERTS; ++e) {
            float v = sp[e];
            if (v > p0)      { p1 = p0; i1 = i0; p0 = v; i0 = e; }
            else if (v > p1) { p1 = v; i1 = e; }
        }
        float s = p0 + p1;
        topk_idx[b * 2 + 0] = i0;  topk_idx[b * 2 + 1] = i1;
        topk_w [b * 2 + 0] = p0 / s;  topk_w [b * 2 + 1] = p1 / s;
        atomicAdd(&counts[i0], 1);
        atomicAdd(&counts[i1], 1);
    }
}

__global__ void scan_kernel(const int* __restrict__ counts,
                            int* __restrict__ padded_off,
                            int* __restrict__ tileExpert,
                            int* __restrict__ numTiles) {
    if (threadIdx.x == 0) {
        int off = 0;
        for (int e = 0; e < N_EXPERTS; ++e) {
            padded_off[e] = off;
            int pc = (counts[e] + MTILE - 1) / MTILE;
            for (int t = 0; t < pc; ++t) tileExpert[off / MTILE + t] = e;
            off += pc * MTILE;
        }
        padded_off[N_EXPERTS] = off;
        numTiles[0] = off / MTILE;
    }
}

__global__ void fill_kernel(int* __restrict__ perm_token) {
    int i = blockIdx.x * blockDim.x + threadIdx.x;
    if (i < CAP) perm_token[i] = -1;
}

__global__ void scatter_kernel(const int* __restrict__ topk_idx,
                               const int* __restrict__ padded_off,
                               int* __restrict__ cursor,
                               int* __restrict__ perm_token,
                               int* __restrict__ pos_assign) {
    int i = blockIdx.x * blockDim.x + threadIdx.x;
    if (i >= BATCH * TOP_K) return;
    int e = topk_idx[i];
    int pos = atomicAdd(&cursor[e], 1);
    int slot = padded_off[e] + pos;
    perm_token[slot] = i >> 1;
    pos_assign[i] = slot;
}

// copy x rows into zero-padded permuted buffer [CAP][XG_STRIDE]
__global__ void gather_kernel(const int* __restrict__ perm_token,
                              const float* __restrict__ x,
                              float* __restrict__ xg) {
    int r = blockIdx.x;
    int t = perm_token[r];
    float* dst = xg + (size_t)r * XG_STRIDE;
    if (t < 0) {
        for (int k = threadIdx.x; k < XG_STRIDE; k += blockDim.x) dst[k] = 0.0f;
    } else {
        const float* src = x + (size_t)t * IN_DIM;
        for (int k = threadIdx.x; k < XG_STRIDE; k += blockDim.x)
            dst[k] = (k < IN_DIM) ? src[k] : 0.0f;
    }
}

// ---------------- WMMA GEMM: Y[m,n] = gelu(sum_k A[m,k]*W[e][n,k] + bias[e][n]) ----------------
// 256 threads = 8 waves. Block tile 64(M) x 256(N). Wave tile 32x64 via 2x4 f32 16x16x4 WMMA accums.
// K%4==0 path: double-buffered LDS filled by GLOBAL_LOAD_ASYNC_TO_LDS_B128 (ASYNCcnt).
// Other K (proj, K=267): single-buffer synchronous staging with per-element guards.

template <int KDIM, int ASTRIDE>
__launch_bounds__(256)
__global__ void gemm_gelu_kernel(const float* __restrict__ A,
                                 const float* __restrict__ W,
                                 const float* __restrict__ bias,
                                 float* __restrict__ Y,
                                 const int* __restrict__ tileExpert,
                                 const int* __restrict__ numTiles,
                                 size_t wExpertStride, size_t bExpertStride) {
    constexpr bool ASYNC = ((KDIM % 4) == 0) && (HAS_ASYNC_LDS != 0);
    constexpr int NBUF = ASYNC ? 2 : 1;
    constexpr int NCH = (KDIM + KCHUNK - 1) / KCHUNK;

    int t = blockIdx.y;
    if (t >= numTiles[0]) return;
    int e = tileExpert[t];
    const float* Wb = W + (size_t)e * wExpertStride;
    const float* Bb = bias + (size_t)e * bExpertStride;
    int row0 = t * MTILE;
    int nblk = blockIdx.x * 256;

    __shared__ __align__(16) float ldsA[NBUF][MTILE * LDS_STRIDE];
    __shared__ __align__(16) float ldsB[NBUF][256 * LDS_STRIDE];

    int tid = threadIdx.x;
    int lane = tid & 31;
    int wv = tid >> 5;
    int wm = wv & 1;          // 2 M-waves
    int wn = wv >> 1;         // 4 N-waves
    int half = lane >> 4;     // K-half select per WMMA layout
    int l16 = lane & 15;

    v8f acc[2][4];
#pragma unroll
    for (int i = 0; i < 2; ++i)
#pragma unroll
        for (int j = 0; j < 4; ++j)
#pragma unroll
            for (int r = 0; r < 8; ++r) acc[i][j][r] = 0.0f;

    // ---- per-chunk async staging: 2 B128 ops (A) + 8 B128 ops (B) per thread = 10/wave ----
    auto stage_async = [&](int kb, int buf) {
#pragma unroll
        for (int it = 0; it < 2; ++it) {
            int idx = it * 256 + tid;
            int r = idx >> 3, kq = idx & 7;
            async_copy16(A + (size_t)(row0 + r) * ASTRIDE + kb + kq * 4,
                         &ldsA[buf][r * LDS_STRIDE + kq * 4]);
        }
#pragma unroll
        for (int it = 0; it < 8; ++it) {
            int idx = it * 256 + tid;
            int n = idx >> 3, kq = idx & 7;
            async_copy16(Wb + (size_t)(nblk + n) * KDIM + kb + kq * 4,
                         &ldsB[buf][n * LDS_STRIDE + kq * 4]);
        }
    };

    // ---- 8 k-steps of K=4: A halves carry k+0..1 / k+2..3, B symmetric ----
    auto compute = [&](int buf) {
#pragma unroll
        for (int ks = 0; ks < KCHUNK; ks += 4) {
            int kk = ks + half * 2;
            v2f a0 = *(const v2f*)&ldsA[buf][(wm * 32 + l16) * LDS_STRIDE + kk];
            v2f a1 = *(const v2f*)&ldsA[buf][(wm * 32 + 16 + l16) * LDS_STRIDE + kk];
#pragma unroll
            for (int ni = 0; ni < 4; ++ni) {
                v2f b = *(const v2f*)&ldsB[buf][(wn * 64 + ni * 16 + l16) * LDS_STRIDE + kk];
                acc[0][ni] = __builtin_amdgcn_wmma_f32_16x16x4_f32(
                    false, a0, false, b, (short)0, acc[0][ni], false, false);
                acc[1][ni] = __builtin_amdgcn_wmma_f32_16x16x4_f32(
                    false, a1, false, b, (short)0, acc[1][ni], false, false);
            }
        }
    };

    if constexpr (ASYNC) {
        stage_async(0, 0);
        for (int ch = 0; ch < NCH; ++ch) {
            int cur = ch & 1;
            if (ch + 1 < NCH) {
                stage_async((ch + 1) * KCHUNK, cur ^ 1);
                WAIT_ASYNC(10);   // retire current buffer's 10 ops (in-order completion)
            } else {
                WAIT_ASYNC(0);
            }
            __syncthreads();      // publish all waves' tiles
            compute(cur);
            __syncthreads();      // everyone done reading before next prefetch overwrites
        }
    } else {
        for (int ch = 0; ch < NCH; ++ch) {
            int kb = ch * KCHUNK;
            __syncthreads();
            // A: buffers are zero-padded to ASTRIDE (multiple of KCHUNK) -> unguarded float4
#pragma unroll
            for (int it = 0; it < 2; ++it) {
                int idx = it * 256 + tid;
                int r = idx >> 3, kq = idx & 7;
                float4 v = *(const float4*)(A + (size_t)(row0 + r) * ASTRIDE + kb + kq * 4);
                *(float4*)&ldsA[0][r * LDS_STRIDE + kq * 4] = v;
            }
            if constexpr ((KDIM % 4) == 0) {
#pragma unroll
                for (int it = 0; it < 8; ++it) {
                    int idx = it * 256 + tid;
                    int n = idx >> 3, kq = idx & 7;
                    float4 v = *(const float4*)(Wb + (size_t)(nblk + n) * KDIM + kb + kq * 4);
                    *(float4*)&ldsB[0][n * LDS_STRIDE + kq * 4] = v;
                }
            } else {
#pragma unroll
                for (int it = 0; it < 32; ++it) {
                    int idx = it * 256 + tid;
                    int n = idx >> 5, kk = idx & 31;
                    int kg = kb + kk;
                    ldsB[0][n * LDS_STRIDE + kk] =
                        (kg < KDIM) ? Wb[(size_t)(nblk + n) * KDIM + kg] : 0.0f;
                }
            }
            __syncthreads();
            compute(0);
        }
    }

    // epilogue: bias + exact-erf gelu. C layout: VGPR r -> M = r (+8 for upper half), N = lane
#pragma unroll
    for (int mi = 0; mi < 2; ++mi) {
        int mbase = row0 + wm * 32 + mi * 16 + half * 8;
#pragma unroll
        for (int ni = 0; ni < 4; ++ni) {
            int ng = nblk + wn * 64 + ni * 16 + l16;
            float bv = Bb[ng];
#pragma unroll
            for (int r = 0; r < 8; ++r) {
                float v = acc[mi][ni][r] + bv;
                Y[(size_t)(mbase + r) * HID + ng] = gelu_erf(v);
            }
        }
    }
}

// ---------------- classifier head + combine ----------------

__global__ void cls_kernel(const float* __restrict__ h,
                           const float* __restrict__ cw,
                           const float* __restrict__ cb,
                           const int* __restrict__ tileExpert,
                           const int* __restrict__ numTiles,
                           float* __restrict__ clsout) {
    int wv = threadIdx.x >> 5, lane = threadIdx.x & 31;
    int row = blockIdx.x * 8 + wv;
    if (row >= numTiles[0] * MTILE) return;
    int e = tileExpert[row / MTILE];
    const float* hr = h + (size_t)row * HID;
    for (int c = 0; c < N_CLASSES; ++c) {
        const float* wr = cw + ((size_t)e * N_CLASSES + c) * HID;
        float s = 0.0f;
        for (int k = lane; k < HID; k += 32) s += hr[k] * wr[k];
        s = wave_sum32(s);
        if (lane == 0) clsout[row * N_CLASSES + c] = s + cb[e * N_CLASSES + c];
    }
}

__global__ void combine_kernel(const float* __restrict__ clsout,
                               const float* __restrict__ topk_w,
                               const int* __restrict__ pos_assign,
                               float* __restrict__ out) {
    int i = blockIdx.x * blockDim.x + threadIdx.x;
    if (i >= BATCH * N_CLASSES) return;
    int b = i / N_CLASSES, c = i % N_CLASSES;
    float r = 0.0f;
    for (int k = 0; k < TOP_K; ++k) {
        int slot = pos_assign[b * TOP_K + k];
        r += topk_w[b * TOP_K + k] * clsout[slot * N_CLASSES + c];
    }
    out[i] = r;
}

// ---------------- launch ----------------

extern "C" void kernel_launch(void* const* d_in, const int* in_sizes, int n_in,
                              void* d_out, int out_size, void* d_ws, size_t ws_size,
                              hipStream_t stream) {
    const float* x  = (const float*)d_in[0];
    const float* rw = (const float*)d_in[1];
    const float* rb = (const float*)d_in[2];
    const float* pw = (const float*)d_in[3];
    const float* pb = (const float*)d_in[4];
    const float* lw = (const float*)d_in[5];
    const float* lb = (const float*)d_in[6];
    const float* cw = (const float*)d_in[7];
    const float* cb = (const float*)d_in[8];
    float* out = (float*)d_out;

    char* ws = (char*)d_ws;
    size_t off = 0;
    auto alloc = [&](size_t bytes) -> char* {
        char* p = ws + off;
        off = (off + bytes + 255) & ~(size_t)255;
        return p;
    };
    int*   counts     = (int*)  alloc(N_EXPERTS * sizeof(int));
    int*   cursor     = (int*)  alloc(N_EXPERTS * sizeof(int));
    int*   padded_off = (int*)  alloc((N_EXPERTS + 1) * sizeof(int));
    int*   numTiles   = (int*)  alloc(sizeof(int));
    int*   tileExpert = (int*)  alloc(MAX_MTILES * sizeof(int));
    int*   topk_idx   = (int*)  alloc(BATCH * TOP_K * sizeof(int));
    int*   pos_assign = (int*)  alloc(BATCH * TOP_K * sizeof(int));
    float* topk_w     = (float*)alloc(BATCH * TOP_K * sizeof(float));
    int*   perm_token = (int*)  alloc(CAP * sizeof(int));
    float* clsout     = (float*)alloc((size_t)CAP * N_CLASSES * sizeof(float));
    float* xg         = (float*)alloc((size_t)CAP * XG_STRIDE * sizeof(float));
    float* h0         = (float*)alloc((size_t)CAP * HID * sizeof(float));
    float* h1         = (float*)alloc((size_t)CAP * HID * sizeof(float));

    init_kernel<<<1, 64, 0, stream>>>(counts, cursor);
    router_kernel<<<BATCH, 32, 0, stream>>>(x, rw, rb, topk_idx, topk_w, counts);
    scan_kernel<<<1, 32, 0, stream>>>(counts, padded_off, tileExpert, numTiles);
    fill_kernel<<<(CAP + 255) / 256, 256, 0, stream>>>(perm_token);
    scatter_kernel<<<(BATCH * TOP_K + 255) / 256, 256, 0, stream>>>(
        topk_idx, padded_off, cursor, perm_token, pos_assign);
    gather_kernel<<<CAP, 64, 0, stream>>>(perm_token, x, xg);

    dim3 gg(HID / 256, MAX_MTILES);
    // proj: K = 267 (A buffer zero-padded to 288), W stride per expert = HID*IN_DIM
    gemm_gelu_kernel<IN_DIM, XG_STRIDE><<<gg, 256, 0, stream>>>(
        xg, pw, pb, h0, tileExpert, numTiles,
        (size_t)HID * IN_DIM, (size_t)HID);
    // 4 hidden layers, ping-pong h0/h1
    const float* hin = h0; float* hout = h1;
    for (int l = 0; l < N_LAYERS; ++l) {
        gemm_gelu_kernel<HID, HID><<<gg, 256, 0, stream>>>(
            hin, lw + (size_t)l * HID * HID, lb + (size_t)l * HID,
            hout, tileExpert, numTiles,
            (size_t)N_LAYERS * HID * HID, (size_t)N_LAYERS * HID);
        const float* tmp = hout; hout = (float*)hin; hin = tmp;
    }
    // after 4 layers the final activations are back in h0 (hin points at them)
    cls_kernel<<<CAP / 8, 256, 0, stream>>>(hin, cw, cb, tileExpert, numTiles, clsout);
    combine_kernel<<<(BATCH * N_CLASSES + 255) / 256, 256, 0, stream>>>(
        clsout, topk_w, pos_assign, out);
    (void)in_sizes; (void)n_in; (void)out_size; (void)ws_size;
}